// NodeSepDropoutLayer_72164040508019
// MI455X (gfx1250) — compile-verified
//
#include <hip/hip_runtime.h>

// ---------------------------------------------------------------------------
// NodeSepDropoutLayer collapsed form:
//   The row-scan adds exactly 256/row and cannot reach ntz=20,132,659 in
//   65536 rows -> data-independent: reached==false, remaining_p = 3355443/2^24.
//   take_t = isin(label,tcls) && attack && row_rand<=0.9
//     take_t -> keep cols [256,512) (chunks 2,3), zero cols [0,256)
//     else   -> keep cols [0,256)  (chunks 0,1), zero cols [256,512)
//   kept elem: out = (residual_rand > remaining_p) ? input*2.5f : 0
//
// Wave-per-4-rows layout: lane l covers float4 {l, l+32, l+64, l+96} of each
// row, so chunk index == column quarter. Keep/zero choice and all address
// stepping are scalar; zero half skips ALL reads (1.5x traffic cut).
// ---------------------------------------------------------------------------

typedef float v4f __attribute__((ext_vector_type(4)));

constexpr int   ROWS     = 65536;
constexpr int   COLS     = 512;
constexpr int   F4_ROW   = COLS / 4;                 // 128 float4 per row
constexpr float REM_P    = 3355443.0f / 16777216.0f; // exact, matches JAX f32
constexpr float SCALE    = 2.5f;                     // f32(1/(1-0.6))
constexpr int   BLOCK    = 256;                      // 8 waves
constexpr int   RPW      = 4;                        // rows per wave
constexpr int   ROWS_BLK = (BLOCK / 32) * RPW;       // 32 rows per block
constexpr int   GRID     = ROWS / ROWS_BLK;          // 2048 blocks

__global__ __launch_bounds__(BLOCK)
void NodeSepDropoutLayer_kernel(const v4f*  __restrict__ in4,
                                const int*  __restrict__ labels,
                                const int*  __restrict__ tcls,
                                const int*  __restrict__ attackp,
                                const float* __restrict__ row_rand,
                                const v4f*  __restrict__ rr4,
                                v4f*        __restrict__ out4)
{
    const int lane = threadIdx.x & 31;
    // wave id is uniform by construction; force SGPR so all row addressing
    // and keep/zero decisions stay scalar (s_load + s_cmp + s_cselect).
    const int wave = __builtin_amdgcn_readfirstlane(
        (int)(blockIdx.x * (BLOCK / 32) + (threadIdx.x >> 5)));
    const int row0 = wave * RPW;

    // Grid-uniform parameters: one scalar fetch per wave.
    const int t0     = tcls[0];
    const int t1     = tcls[1];
    const int attack = attackp[0];

    const v4f zero = {0.0f, 0.0f, 0.0f, 0.0f};

#pragma unroll
    for (int r = 0; r < RPW; ++r) {
        const int row = row0 + r;
        const int lab = labels[row];                        // s_load (scalar$)
        const bool take_t = (attack != 0) &&
                            ((lab == t0) || (lab == t1)) &&
                            (row_rand[row] <= 0.9f);        // s_cmp_le_f32

        // chunk i covers cols [i*128,(i+1)*128); keep {2,3} iff take_t.
        const int keepc = take_t ? 2 : 0;   // scalar select
        const int zeroc = 2 - keepc;

        const unsigned rowbase = (unsigned)row * F4_ROW + (unsigned)lane;

#pragma unroll
        for (int i = 0; i < 2; ++i) {
            const unsigned k = rowbase + (unsigned)(keepc + i) * 32u;
            // Streamed once, 268 MB total > 192 MB L2 -> non-temporal b128.
            v4f a = __builtin_nontemporal_load(&in4[k]);
            v4f q = __builtin_nontemporal_load(&rr4[k]);
            v4f o;
            o.x = (q.x > REM_P) ? a.x * SCALE : 0.0f;
            o.y = (q.y > REM_P) ? a.y * SCALE : 0.0f;
            o.z = (q.z > REM_P) ? a.z * SCALE : 0.0f;
            o.w = (q.w > REM_P) ? a.w * SCALE : 0.0f;
            __builtin_nontemporal_store(o, &out4[k]);

            // Dropped half-row: no reads, just 128-bit NT zero stores.
            const unsigned z = rowbase + (unsigned)(zeroc + i) * 32u;
            __builtin_nontemporal_store(zero, &out4[z]);
        }
    }
}

extern "C" void kernel_launch(void* const* d_in, const int* in_sizes, int n_in,
                              void* d_out, int out_size, void* d_ws, size_t ws_size,
                              hipStream_t stream)
{
    (void)in_sizes; (void)n_in; (void)out_size; (void)d_ws; (void)ws_size;

    const v4f*   in4      = (const v4f*)  d_in[0];  // input         [65536,512] f32
    const int*   labels   = (const int*)  d_in[1];  // labels        [65536]     i32
    const int*   tcls     = (const int*)  d_in[2];  // target_classes [2]        i32
    const int*   attackp  = (const int*)  d_in[3];  // start_attack  scalar      i32
    const float* row_rand = (const float*)d_in[4];  // row_rand      [65536]     f32
    const v4f*   rr4      = (const v4f*)  d_in[5];  // residual_rand [65536,512] f32
    v4f*         out4     = (v4f*)        d_out;

    NodeSepDropoutLayer_kernel<<<GRID, BLOCK, 0, stream>>>(
        in4, labels, tcls, attackp, row_rand, rr4, out4);
}